// LinearGaussianELBO_52037823758956
// MI455X (gfx1250) — compile-verified
//
#include <hip/hip_runtime.h>
#include <math.h>

typedef __attribute__((ext_vector_type(2))) float v2f;
typedef __attribute__((ext_vector_type(8))) float v8f;

#define NW 8                       // waves per block (256 threads / 32)
#define LOG2PI_F 1.83787706640934548356f

// ---------- WMMA helpers (fp32 16x16x4, chained to 16x16x16) ----------

__device__ __forceinline__ v8f wmma4(v2f a, v2f b, v8f c) {
  // 8 args: (neg_a, A, neg_b, B, c_mod, C, reuse_a, reuse_b)
  return __builtin_amdgcn_wmma_f32_16x16x4_f32(false, a, false, b, (short)0, c,
                                               false, false);
}

// A-matrix 16x4 fp32 fragment from row-major 16x16: lanes 0-15 hold rows,
// VGPR0 = K{k0,k0+2}, VGPR1 = K{k0+1,k0+3} split across lane halves.
__device__ __forceinline__ v2f ld_afrag(const float* M, int k0, int lane) {
  int r = lane & 15;
  int kk = k0 + ((lane >> 4) << 1);
  v2f a;
  a.x = M[r * 16 + kk];
  a.y = M[r * 16 + kk + 1];
  return a;
}

// B-matrix 4x16 fp32 fragment from row-major 16x16 (rows striped over lanes).
__device__ __forceinline__ v2f ld_bfrag(const float* M, int k0, int lane) {
  int c = lane & 15;
  int kk = k0 + ((lane >> 4) << 1);
  v2f b;
  b.x = M[kk * 16 + c];
  b.y = M[(kk + 1) * 16 + c];
  return b;
}

// Full 16x16 fp32 GEMM per wave: D = A(16x16) @ B(16x16) via 4x WMMA k-slices.
__device__ __forceinline__ v8f mm_wmma(const float* A, const float* B, int lane) {
  v8f acc = {0.f, 0.f, 0.f, 0.f, 0.f, 0.f, 0.f, 0.f};
#pragma unroll
  for (int k0 = 0; k0 < 16; k0 += 4)
    acc = wmma4(ld_afrag(A, k0, lane), ld_bfrag(B, k0, lane), acc);
  return acc;
}

// Store C/D fragment (VGPR r: lanes0-15 -> row r, lanes16-31 -> row r+8).
__device__ __forceinline__ void st_cd(float* Dst, v8f c, int lane) {
  int col = lane & 15, rb = (lane >> 4) << 3;
#pragma unroll
  for (int r = 0; r < 8; r++) Dst[(rb + r) * 16 + col] = c[r];
}

__device__ __forceinline__ float wave_sum(float v) {
#pragma unroll
  for (int o = 16; o > 0; o >>= 1) v += __shfl_xor(v, o, 32);
  return v;
}

// ---------- 256-thread cooperative 16x16 helpers ----------

__device__ __forceinline__ void mm16(float* Dm, const float* A, const float* B, int tid) {
  int i = tid >> 4, j = tid & 15;
  float s = 0.f;
#pragma unroll
  for (int k = 0; k < 16; k++) s = fmaf(A[i * 16 + k], B[k * 16 + j], s);
  Dm[tid] = s;
}
__device__ __forceinline__ void mm16T(float* Dm, const float* A, const float* B, int tid) {
  int i = tid >> 4, j = tid & 15;  // D = A @ B^T
  float s = 0.f;
#pragma unroll
  for (int k = 0; k < 16; k++) s = fmaf(A[i * 16 + k], B[j * 16 + k], s);
  Dm[tid] = s;
}
__device__ __forceinline__ void mTm16(float* Dm, const float* A, const float* B, int tid) {
  int i = tid >> 4, j = tid & 15;  // D = A^T @ B
  float s = 0.f;
#pragma unroll
  for (int k = 0; k < 16; k++) s = fmaf(A[k * 16 + i], B[k * 16 + j], s);
  Dm[tid] = s;
}

// Gauss-Jordan inverse of SPD-ish 16x16 (no pivoting). Destroys M, writes Inv.
// sc[0]=pivot scratch, sc[1]=log|det(M)| (sum of log pivots).
__device__ void inv16(float* M, float* Inv, float* colf, float* sc, int tid) {
  { int i = tid >> 4, j = tid & 15; Inv[tid] = (i == j) ? 1.f : 0.f; }
  if (tid == 0) sc[1] = 0.f;
  __syncthreads();
  for (int p = 0; p < 16; p++) {
    if (tid == 0) {
      float pv = M[p * 16 + p];
      sc[0] = 1.f / pv;
      sc[1] += logf(fabsf(pv));
    }
    __syncthreads();
    if (tid < 16) { M[p * 16 + tid] *= sc[0]; Inv[p * 16 + tid] *= sc[0]; }
    __syncthreads();
    if (tid < 16) colf[tid] = (tid == p) ? 0.f : M[tid * 16 + p];
    __syncthreads();
    {
      int i = tid >> 4, j = tid & 15;
      float f = colf[i];
      float mp = M[p * 16 + j], ip = Inv[p * 16 + j];
      if (i != p) { M[tid] -= f * mp; Inv[tid] -= f * ip; }
    }
    __syncthreads();
  }
}

__device__ __forceinline__ float clgf(float logdet) {
  return -0.5f * (16.f * LOG2PI_F + logdet);
}

// Kalman measurement update under the v-model.
// mean (vav) / cov (s2) are inputs; writes mfv, Pfm. Clobbers s0,s1,s3,gjM,gjI.
__device__ void kalman_meas_update(const float* y, float* mean, float* cov,
                                   const float* vHm, const float* vRm,
                                   const float* vebv, float* s0, float* s1,
                                   float* s3, float* gjM, float* gjI,
                                   float* colf, float* sc, float* vbv,
                                   float* mfv, float* Pfm, int tid) {
  mm16(s0, vHm, cov, tid); __syncthreads();           // H @ P
  mm16T(s1, s0, vHm, tid); __syncthreads();           // (H P) H^T
  gjM[tid] = s1[tid] + vRm[tid]; __syncthreads();     // S
  inv16(gjM, gjI, colf, sc, tid);                     // Sinv
  mm16T(s3, cov, vHm, tid); __syncthreads();          // P H^T
  mm16(s0, s3, gjI, tid); __syncthreads();            // K
  if (tid < 16) {
    float s = 0.f;
#pragma unroll
    for (int j = 0; j < 16; j++) s = fmaf(vHm[tid * 16 + j], mean[j], s);
    vbv[tid] = y[tid] - s - vebv[tid];                // innovation
  }
  __syncthreads();
  if (tid < 16) {
    float s = 0.f;
#pragma unroll
    for (int j = 0; j < 16; j++) s = fmaf(s0[tid * 16 + j], vbv[j], s);
    mfv[tid] = mean[tid] + s;
  }
  mm16(s1, s0, vHm, tid); __syncthreads();            // K H
  s3[tid] = (((tid >> 4) == (tid & 15)) ? 1.f : 0.f) - s1[tid];
  __syncthreads();
  mm16(Pfm, s3, cov, tid); __syncthreads();           // (I - K H) P
}

// ---------- main persistent kernel ----------

__global__ void __launch_bounds__(256)
lg_elbo_kernel(const float* __restrict__ obs, const float* __restrict__ prior_mean_g,
               const float* __restrict__ prior_cov_g, const float* __restrict__ trans_w_g,
               const float* __restrict__ trans_b_g, const float* __restrict__ trans_cov_g,
               const float* __restrict__ em_w_g, const float* __restrict__ em_b_g,
               const float* __restrict__ em_cov_g, const float* __restrict__ v_prior_mean_g,
               const float* __restrict__ v_prior_cov_g, const float* __restrict__ v_trans_w_g,
               const float* __restrict__ v_trans_b_g, const float* __restrict__ v_trans_cov_g,
               const float* __restrict__ v_em_w_g, const float* __restrict__ v_em_b_g,
               const float* __restrict__ v_em_cov_g, float* __restrict__ out,
               float* __restrict__ Aws, float* __restrict__ bws, int T) {
  const int tid = threadIdx.x;
  const int lane = tid & 31;
  const int wave = tid >> 5;

  __shared__ float prec_em[256], prec_tr[256], prec_pr[256], vPm[256];
  __shared__ float Pfm[256], bcov[256], bAm[256], ctm[256];
  __shared__ float Wem[256], Wtr[256], vWm[256], vHm[256], vQm[256], vRm[256];
  __shared__ float s0[256], s1[256], s2[256], s3[256];
  __shared__ float gjM[256], gjI[256];
  __shared__ float wtile[NW][256];
  __shared__ float wvec[NW][16];
  __shared__ float mfv[16], bav[16], vav[16], vbv[16];
  __shared__ float tbv[16], ebv[16], vtbv[16], vebv[16];
  __shared__ float colf[16], sc[2];
  __shared__ float tr_acc, acc2, ld_back, c_acc;
  __shared__ float em_ld, tr_ld, pr_ld;

  // ---- setup: cache weights, compute precisions + log-dets ----
  Wem[tid] = em_w_g[tid];  Wtr[tid] = trans_w_g[tid];
  vWm[tid] = v_trans_w_g[tid]; vHm[tid] = v_em_w_g[tid];
  vQm[tid] = v_trans_cov_g[tid]; vRm[tid] = v_em_cov_g[tid];
  if (tid < 16) {
    tbv[tid] = trans_b_g[tid];  ebv[tid] = em_b_g[tid];
    vtbv[tid] = v_trans_b_g[tid]; vebv[tid] = v_em_b_g[tid];
  }
  gjM[tid] = em_cov_g[tid]; __syncthreads();
  inv16(gjM, gjI, colf, sc, tid);
  prec_em[tid] = gjI[tid]; if (tid == 0) em_ld = sc[1];
  __syncthreads();
  gjM[tid] = trans_cov_g[tid]; __syncthreads();
  inv16(gjM, gjI, colf, sc, tid);
  prec_tr[tid] = gjI[tid]; if (tid == 0) tr_ld = sc[1];
  __syncthreads();
  gjM[tid] = prior_cov_g[tid]; __syncthreads();
  inv16(gjM, gjI, colf, sc, tid);
  prec_pr[tid] = gjI[tid]; if (tid == 0) pr_ld = sc[1];
  __syncthreads();
  gjM[tid] = v_trans_cov_g[tid]; __syncthreads();
  inv16(gjM, gjI, colf, sc, tid);
  vPm[tid] = gjI[tid];
  __syncthreads();

  // quad forms 0 and 1 (Om is implicit: even n -> em_prec, n==1 -> prior, odd -> trans)
  { int i = tid >> 4, j = tid & 15;
    Aws[tid] = Wem[tid];
    Aws[256 + tid] = (i == j) ? 1.f : 0.f; }
  if (tid < 16) { bws[tid] = ebv[tid] - obs[tid]; bws[16 + tid] = -prior_mean_g[tid]; }

  // init filter with first observation
  if (tid < 16) vav[tid] = v_prior_mean_g[tid];
  s2[tid] = v_prior_cov_g[tid];
  if (tid == 0) c_acc = clgf(pr_ld) + clgf(em_ld);
  __syncthreads();
  kalman_meas_update(obs, vav, s2, vHm, vRm, vebv, s0, s1, s3, gjM, gjI, colf,
                     sc, vbv, mfv, Pfm, tid);

  // ---- sequential scan over time ----
  for (int t = 1; t < T; ++t) {
    const float* y = obs + t * 16;
    const int count = 2 * t;  // active quad forms before appending

    // filt_prec = inv(Pf) -> s0
    gjM[tid] = Pfm[tid]; __syncthreads();
    inv16(gjM, gjI, colf, sc, tid);
    s0[tid] = gjI[tid]; __syncthreads();

    // ct = vW^T @ v_trans_prec
    mTm16(ctm, vWm, vPm, tid); __syncthreads();

    // back_prec = ct @ vW + filt_prec ; back_cov = inv(back_prec)
    { int i = tid >> 4, j = tid & 15; float s = 0.f;
#pragma unroll
      for (int k = 0; k < 16; k++) s = fmaf(ctm[i * 16 + k], vWm[k * 16 + j], s);
      gjM[tid] = s + s0[tid]; }
    __syncthreads();
    inv16(gjM, gjI, colf, sc, tid);
    bcov[tid] = gjI[tid]; if (tid == 0) ld_back = -sc[1];
    __syncthreads();

    mm16(bAm, bcov, ctm, tid); __syncthreads();      // back_A

    if (tid < 16) {                                  // va = filt_prec@mf - ct@v_trans_b
      float s = 0.f;
#pragma unroll
      for (int j = 0; j < 16; j++)
        s += s0[tid * 16 + j] * mfv[j] - ctm[tid * 16 + j] * vtbv[j];
      vav[tid] = s;
    }
    if (tid == 0) { tr_acc = 0.f; acc2 = 0.f; }
    __syncthreads();
    if (tid < 16) {                                  // back_a = back_cov @ va
      float s = 0.f;
#pragma unroll
      for (int j = 0; j < 16; j++) s = fmaf(bcov[tid * 16 + j], vav[j], s);
      bav[tid] = s;
    }
    __syncthreads();

    // ---- batched quad-form pass: per-wave WMMA GEMMs ----
    for (int n = wave; n < count; n += NW) {
      float* An = Aws + (size_t)n * 256;
      v8f y1 = mm_wmma(An, bcov, lane);              // Y = A @ back_cov
      st_cd(&wtile[wave][0], y1, lane);
      asm volatile("" ::: "memory");                 // per-wave DS ops are in-order
      const float* om = (n == 1) ? prec_pr : ((n & 1) ? prec_tr : prec_em);
      v8f z = mm_wmma(om, &wtile[wave][0], lane);    // Z = Prec @ Y
      int col = lane & 15, rb = (lane >> 4) << 3;
      float part = 0.f;                              // tr += Z .* A  (scaled -0.5 later)
#pragma unroll
      for (int r = 0; r < 8; r++) part = fmaf(z[r], An[(rb + r) * 16 + col], part);
      float bn = 0.f;                                // old-A matvec for b update
      if (lane < 16) {
#pragma unroll
        for (int j = 0; j < 16; j++) bn = fmaf(An[lane * 16 + j], bav[j], bn);
      }
      v8f anew = mm_wmma(An, bAm, lane);             // A_new = A @ back_A
      float wsum = wave_sum(part);
      if (lane == 0) atomicAdd(&tr_acc, wsum);
      if (lane < 16) bws[n * 16 + lane] += bn;
      st_cd(An, anew, lane);
    }
    __threadfence();
    __syncthreads();

    // append new quad forms at count, count+1
    mm16(s2, Wtr, bAm, tid); __syncthreads();
    { int i = tid >> 4, j = tid & 15;
      Aws[(size_t)count * 256 + tid] = Wem[tid];
      Aws[(size_t)(count + 1) * 256 + tid] = s2[tid] - ((i == j) ? 1.f : 0.f); }
    if (tid < 16) {
      bws[count * 16 + tid] = ebv[tid] - y[tid];
      float s = 0.f;
#pragma unroll
      for (int j = 0; j < 16; j++) s = fmaf(Wtr[tid * 16 + j], bav[j], s);
      bws[(count + 1) * 16 + tid] = s + tbv[tid];
    }
    __syncthreads();

    // trace(trans_prec @ trans_w @ back_cov @ trans_w^T)
    mm16(s1, Wtr, bcov, tid); __syncthreads();
    mm16(s2, prec_tr, s1, tid); __syncthreads();
    atomicAdd(&acc2, s2[tid] * Wtr[tid]);
    __syncthreads();

    if (tid == 0) {
      c_acc += clgf(em_ld) + clgf(tr_ld) - 0.5f * acc2 - clgf(ld_back) + 8.0f -
               0.5f * tr_acc;
    }
    __syncthreads();

    // Kalman predict + update
    if (tid < 16) {
      float s = 0.f;
#pragma unroll
      for (int j = 0; j < 16; j++) s = fmaf(vWm[tid * 16 + j], mfv[j], s);
      vav[tid] = s + vtbv[tid];                      // predicted mean
    }
    mm16(s0, vWm, Pfm, tid); __syncthreads();
    mm16T(s1, s0, vWm, tid); __syncthreads();
    s2[tid] = s1[tid] + vQm[tid];                    // predicted cov
    __syncthreads();
    kalman_meas_update(y, vav, s2, vHm, vRm, vebv, s0, s1, s3, gjM, gjI, colf,
                       sc, vbv, mfv, Pfm, tid);
  }

  // ---- epilogue ----
  const int Nq = 2 * T;
  gjM[tid] = Pfm[tid]; __syncthreads();
  inv16(gjM, gjI, colf, sc, tid);                    // log det(Pf) from pivots
  if (tid == 0) { c_acc -= clgf(sc[1]); tr_acc = 0.f; }
  __syncthreads();

  for (int n = wave; n < Nq; n += NW) {
    const float* An = Aws + (size_t)n * 256;
    v8f y1 = mm_wmma(An, Pfm, lane);                 // Y = A @ Pf
    st_cd(&wtile[wave][0], y1, lane);
    asm volatile("" ::: "memory");
    const float* om = (n == 1) ? prec_pr : ((n & 1) ? prec_tr : prec_em);
    v8f z = mm_wmma(om, &wtile[wave][0], lane);
    int col = lane & 15, rb = (lane >> 4) << 3;
    float part = 0.f;                                // tr_f: Z .* A^T (reference's 'nli')
#pragma unroll
    for (int r = 0; r < 8; r++) part = fmaf(z[r], An[col * 16 + rb + r], part);
    if (lane < 16) {                                 // v = A@mf + b
      float s = bws[n * 16 + lane];
#pragma unroll
      for (int j = 0; j < 16; j++) s = fmaf(An[lane * 16 + j], mfv[j], s);
      wvec[wave][lane] = s;
    }
    asm volatile("" ::: "memory");
    float part2 = 0.f;                               // quad = v^T Prec v
    if (lane < 16) {
      float w = 0.f;
#pragma unroll
      for (int j = 0; j < 16; j++) w = fmaf(om[lane * 16 + j], wvec[wave][j], w);
      part2 = wvec[wave][lane] * w;
    }
    float wsum = wave_sum(part + part2);
    if (lane == 0) atomicAdd(&tr_acc, wsum);
  }
  __syncthreads();
  if (tid == 0) out[0] = c_acc - 0.5f * tr_acc + 8.0f;  // half_dz = 8
}

extern "C" void kernel_launch(void* const* d_in, const int* in_sizes, int n_in,
                              void* d_out, int out_size, void* d_ws, size_t ws_size,
                              hipStream_t stream) {
  (void)n_in; (void)out_size; (void)ws_size;
  const float* obs          = (const float*)d_in[0];
  const float* prior_mean   = (const float*)d_in[1];
  const float* prior_cov    = (const float*)d_in[2];
  const float* trans_w      = (const float*)d_in[3];
  const float* trans_b      = (const float*)d_in[4];
  const float* trans_cov    = (const float*)d_in[5];
  const float* em_w         = (const float*)d_in[6];
  const float* em_b         = (const float*)d_in[7];
  const float* em_cov       = (const float*)d_in[8];
  const float* v_prior_mean = (const float*)d_in[9];
  const float* v_prior_cov  = (const float*)d_in[10];
  const float* v_trans_w    = (const float*)d_in[11];
  const float* v_trans_b    = (const float*)d_in[12];
  const float* v_trans_cov  = (const float*)d_in[13];
  const float* v_em_w       = (const float*)d_in[14];
  const float* v_em_b       = (const float*)d_in[15];
  const float* v_em_cov     = (const float*)d_in[16];

  int T = in_sizes[0] / 16;
  int N = 2 * T;
  float* Aws = (float*)d_ws;                 // N x 16 x 16
  float* bws = Aws + (size_t)N * 256;        // N x 16

  lg_elbo_kernel<<<1, 256, 0, stream>>>(
      obs, prior_mean, prior_cov, trans_w, trans_b, trans_cov, em_w, em_b,
      em_cov, v_prior_mean, v_prior_cov, v_trans_w, v_trans_b, v_trans_cov,
      v_em_w, v_em_b, v_em_cov, (float*)d_out, Aws, bws, T);
}